// StridedSparseSelfAttention_64957085384893
// MI455X (gfx1250) — compile-verified
//
#include <hip/hip_runtime.h>
#include <hip/hip_bf16.h>

// ---------------------------------------------------------------------------
// StridedSparseSelfAttention for MI455X (gfx1250, wave32, WMMA).
// EMB=256 HEADS=8 K=8 GADD=2 RADD=2 REGION=64 STRIDE=32 B=4 T=4096
// tp = 128, vs = 48, sel rows total = 512.
// Algorithmic restructure: never materialize full K/V projections.
//   qk_h  = (x_sel @ Wq)_h @ Wk_h^T / 16          (512x256 per head)
//   dot_v = qk_h . x[idx_v]                        (48 per (b,p,h))
//   att   = softmax(weights * dot)
//   ctx_h = sum_v att_v * x[idx_v]                 (512x256 per head)
//   out_sel = concat_h(ctx_h @ Wv_h) @ Wu + bu ; other rows = bu
// All GEMMs use V_WMMA_F32_16X16X4_F32 (f32 in, f32 acc) for fp32 fidelity.
// ---------------------------------------------------------------------------

typedef float v2f __attribute__((ext_vector_type(2)));
typedef float v8f __attribute__((ext_vector_type(8)));

#define EMB   256
#define HEADS 8
#define KK    8
#define TT    4096
#define BB    4
#define TP    128
#define VS    48
#define ROWS  512   // B * TP

// ------------------------- threefry2x32 (JAX-compatible) -------------------
__host__ __device__ inline unsigned tf_rotl(unsigned v, int n) {
  return (v << n) | (v >> (32 - n));
}

__host__ __device__ inline void threefry2x32(unsigned k0, unsigned k1,
                                             unsigned x0, unsigned x1,
                                             unsigned* o0, unsigned* o1) {
  unsigned ks2 = k0 ^ k1 ^ 0x1BD11BDAu;
  unsigned a = x0 + k0, b = x1 + k1;
  const int r0[4] = {13, 15, 26, 6};
  const int r1[4] = {17, 29, 16, 24};
  for (int i = 0; i < 4; i++) { a += b; b = tf_rotl(b, r0[i]); b ^= a; }
  a += k1;  b += ks2 + 1u;
  for (int i = 0; i < 4; i++) { a += b; b = tf_rotl(b, r1[i]); b ^= a; }
  a += ks2; b += k0 + 2u;
  for (int i = 0; i < 4; i++) { a += b; b = tf_rotl(b, r0[i]); b ^= a; }
  a += k0;  b += k1 + 3u;
  for (int i = 0; i < 4; i++) { a += b; b = tf_rotl(b, r1[i]); b ^= a; }
  a += k1;  b += ks2 + 4u;
  for (int i = 0; i < 4; i++) { a += b; b = tf_rotl(b, r0[i]); b ^= a; }
  a += ks2; b += k0 + 5u;
  *o0 = a; *o1 = b;
}

__device__ inline float softplusf(float x) {
  return fmaxf(x, 0.f) + log1pf(expf(-fabsf(x)));
}

__device__ inline float wave_sum(float v) {
  v += __shfl_xor(v, 16, 32);
  v += __shfl_xor(v, 8, 32);
  v += __shfl_xor(v, 4, 32);
  v += __shfl_xor(v, 2, 32);
  v += __shfl_xor(v, 1, 32);
  return v;
}

// ------------------------------ small kernels ------------------------------
__global__ void init_rowmap(int* __restrict__ rm) {
  int r = blockIdx.x * blockDim.x + threadIdx.x;
  if (r < ROWS) {
    int b = r >> 7, p = r & 127;
    rm[r] = b * TT + ((p + 1) * 32 - 1);   // global x row of sel position
  }
}

__global__ void fill_bias(float* __restrict__ out, const float* __restrict__ bu,
                          int total) {
  int i = blockIdx.x * blockDim.x + threadIdx.x;
  if (i < total) out[i] = bu[i & (EMB - 1)];
}

// hidden = relu([x_sel, coord] @ W1 + b1), one block per sel row
__global__ void mlp_hidden(const float* __restrict__ x,
                           const float* __restrict__ W1,
                           const float* __restrict__ b1,
                           const int* __restrict__ rm,
                           float* __restrict__ hidden) {
  __shared__ float inp[EMB + 1];
  int r = blockIdx.x, tid = threadIdx.x;
  const float* xr = x + (size_t)rm[r] * EMB;
  inp[tid] = xr[tid];
  if (tid == 0) inp[EMB] = (float)(r & 127) / (float)TP;
  __syncthreads();
  for (int i = 0; i < 4; i++) {
    int u = tid + 256 * i;
    float acc = b1[u];
    for (int e = 0; e < EMB + 1; e++)
      acc = fmaf(inp[e], W1[(size_t)e * 1024 + u], acc);
    hidden[(size_t)r * 1024 + u] = fmaxf(acc, 0.f);
  }
}

// params = hidden @ W2 + b2 (16 outputs per row); one block (8 waves) per row
__global__ void mlp_params(const float* __restrict__ hidden,
                           const float* __restrict__ W2,
                           const float* __restrict__ b2,
                           float* __restrict__ params) {
  int r = blockIdx.x, tid = threadIdx.x;
  int wave = tid >> 5, lane = tid & 31;
  const float* h = hidden + (size_t)r * 1024;
  for (int oo = 0; oo < 2; ++oo) {
    int o = wave + 8 * oo;
    float s = 0.f;
    for (int i = 0; i < 32; i++) {
      int j = lane + 32 * i;
      s = fmaf(h[j], W2[(size_t)j * 16 + o], s);
    }
    s = wave_sum(s);
    if (lane == 0) params[(size_t)r * 16 + o] = s + b2[o];
  }
}

// means/sigmas -> indices (threefry randint) -> dup/valid -> props -> weights
__global__ void index_weight(const float* __restrict__ params,
                             int* __restrict__ indices,
                             float* __restrict__ weights,
                             unsigned kg0, unsigned kg1,
                             unsigned kr0, unsigned kr1) {
  __shared__ float meanL[KK], sigL[KK], colsum[KK];
  __shared__ int idxL[VS];
  __shared__ float props[VS][KK];
  int r = blockIdx.x, tid = threadIdx.x;
  int p = r & 127;
  int seli = (p + 1) * 32 - 1;
  float self = (float)seli;
  const float* pr = params + (size_t)r * 16;

  if (tid < KK) {
    float mean = self - 3.0f * softplusf(pr[tid]);          // MMULT = 3
    mean = fminf(fmaxf(mean, 0.f), (float)(TT - 1));
    meanL[tid] = mean;
    sigL[tid]  = (softplusf(pr[KK + tid] + 2.0f) + 0.05f) * (float)TT * 0.1f;
    int i0 = (int)floorf(mean);
    idxL[tid * 6 + 0] = i0;
    idxL[tid * 6 + 1] = min(i0 + 1, TT - 1);
  }
  __syncthreads();
  if (tid < 32) {
    int k = tid >> 2, q = tid & 3;
    unsigned hi, lo;
    if (q < 2) {                       // gidx in [0, T): span 4096 pow2 -> lo & 4095
      unsigned j = (unsigned)((r * KK + k) * 2 + q);
      threefry2x32(kg0, kg1, j, 8192u + j, &hi, &lo);
      idxL[k * 6 + 2 + q] = (int)(lo & 4095u);
    } else {                           // ridx in [0, 64): lo & 63, offset by region lo
      unsigned j = (unsigned)((r * KK + k) * 2 + (q - 2));
      threefry2x32(kr0, kr1, j, 8192u + j, &hi, &lo);
      float fl = floorf(meanL[k]);
      float low = fminf(fmaxf(fl - 32.f, 0.f), (float)(TT - 64));
      idxL[k * 6 + 2 + q] = (int)low + (int)(lo & 63u);
    }
  }
  __syncthreads();
  if (tid < VS) {
    int v = tid, iv = idxL[v];
    bool dup = false;
    for (int u = 0; u < v; ++u) dup |= (idxL[u] == iv);
    bool valid = (!dup) && (iv <= seli);
    float fi = (float)iv;
    for (int k = 0; k < KK; k++) {
      float z = (fi - meanL[k]) / sigL[k];
      props[v][k] = valid ? expf(-0.5f * z * z) : 0.f;
    }
  }
  __syncthreads();
  if (tid < KK) {
    float s = 0.f;
    for (int v = 0; v < VS; v++) s += props[v][tid];
    colsum[tid] = s;
  }
  __syncthreads();
  if (tid < VS) {
    float w = 0.f;
    for (int k = 0; k < KK; k++) w += props[tid][k] / colsum[k];
    weights[(size_t)r * VS + tid] = w;
    indices[(size_t)r * VS + tid] = idxL[tid];
  }
}

// dot -> softmax -> ctx. One block per (b,p,h).
__global__ void attn_ctx(const float* __restrict__ x,
                         const float* __restrict__ qk,
                         const int* __restrict__ indices,
                         const float* __restrict__ weights,
                         float* __restrict__ ctx) {
  __shared__ int idxL[VS];
  __shared__ float wL[VS], dotL[VS], attL[VS];
  int blk = blockIdx.x;
  int h = blk & 7, bp = blk >> 3, b = bp >> 7;
  int tid = threadIdx.x, wave = tid >> 5, lane = tid & 31;
  if (tid < VS) {
    idxL[tid] = indices[(size_t)bp * VS + tid];
    wL[tid]   = weights[(size_t)bp * VS + tid];
  }
  __syncthreads();
  const float* qkr = qk + (size_t)bp * 2048 + h * EMB;   // qk already has 1/16
  for (int i = 0; i < 6; i++) {
    int v = wave * 6 + i;
    const float* xr = x + ((size_t)(b * TT + idxL[v])) * EMB;
    float s = 0.f;
    for (int j = 0; j < 8; j++) {
      int e = lane + 32 * j;
      s = fmaf(qkr[e], xr[e], s);
    }
    s = wave_sum(s);
    if (lane == 0) dotL[v] = s;
  }
  __syncthreads();
  if (tid == 0) {
    float m = -1e30f;
    for (int v = 0; v < VS; v++) { float t = wL[v] * dotL[v]; attL[v] = t; m = fmaxf(m, t); }
    float ssum = 0.f;
    for (int v = 0; v < VS; v++) { float e = expf(attL[v] - m); attL[v] = e; ssum += e; }
    float inv = 1.0f / ssum;
    for (int v = 0; v < VS; v++) attL[v] *= inv;
  }
  __syncthreads();
  float acc = 0.f;
  for (int v = 0; v < VS; v++)
    acc = fmaf(attL[v], x[((size_t)(b * TT + idxL[v])) * EMB + tid], acc);
  ctx[((size_t)h * ROWS + bp) * EMB + tid] = acc;
}

// ------------------------- generic WMMA f32 GEMM ---------------------------
// One wave computes one 16x16 tile of C = alpha * A(MxK) @ B(KxN) [+ bias].
// BT: B stored transposed (phys [N][K]). GA: gather A rows via rowmap.
// SC: scatter C rows via rowmap. blockIdx.y offsets (per-head sub-GEMMs).
// f32 WMMA layouts (ISA 7.12.2): A lane l: M=l%16, K=2*(l/16)+j (j=VGPR 0/1);
// C lane l, reg r: M = r + 8*(l/16), N = l%16.
template <bool BT, bool GA, bool SC, bool BIAS>
__global__ __launch_bounds__(256) void wmma_gemm(
    const float* __restrict__ A, int lda, size_t aZ,
    const int* __restrict__ rowmap,
    const float* __restrict__ B, int ldb, size_t bZ,
    float* __restrict__ C, int ldc, size_t cZ,
    const float* __restrict__ bias, float alpha,
    int M, int N, int K) {
  const int tm = M >> 4, tn = N >> 4;
  const int wave = threadIdx.x >> 5;
  const int t = blockIdx.x * (blockDim.x >> 5) + wave;
  if (t >= tm * tn) return;                 // wave-uniform: EXEC stays all-1s
  const int tmi = t % tm, tni = t / tm;
  const int lane = threadIdx.x & 31;
  const int half = lane >> 4, lm = lane & 15;
  const int m0 = tmi * 16, n0 = tni * 16;

  A += aZ * blockIdx.y;
  B += bZ * blockIdx.y;
  C += cZ * blockIdx.y;

  int arow = m0 + lm;
  if (GA) arow = rowmap[arow];
  const float* ap = A + (size_t)arow * lda + 2 * half;
  const float* bp;
  if (BT) bp = B + (size_t)(n0 + lm) * ldb + 2 * half;
  else    bp = B + (size_t)(2 * half) * ldb + (n0 + lm);

  v8f acc = {0.f, 0.f, 0.f, 0.f, 0.f, 0.f, 0.f, 0.f};
  for (int kb = 0; kb < K; kb += 4) {
    v2f a; a[0] = ap[kb]; a[1] = ap[kb + 1];
    v2f b;
    if (BT) { b[0] = bp[kb]; b[1] = bp[kb + 1]; }
    else    { b[0] = bp[(size_t)kb * ldb]; b[1] = bp[(size_t)(kb + 1) * ldb]; }
    acc = __builtin_amdgcn_wmma_f32_16x16x4_f32(
        /*neg_a=*/false, a, /*neg_b=*/false, b,
        /*c_mod=*/(short)0, acc, /*reuse_a=*/false, /*reuse_b=*/false);
  }

  const int col = n0 + lm;
  const float bv = BIAS ? bias[col] : 0.f;
#pragma unroll
  for (int r = 0; r < 8; ++r) {
    int row = m0 + r + 8 * half;
    int crow = SC ? rowmap[row] : row;
    C[(size_t)crow * ldc + col] = acc[r] * alpha + bv;
  }
}

// --------------------------------- launch ----------------------------------
extern "C" void kernel_launch(void* const* d_in, const int* in_sizes, int n_in,
                              void* d_out, int out_size, void* d_ws, size_t ws_size,
                              hipStream_t stream) {
  const float* x  = (const float*)d_in[0];
  const float* Wq = (const float*)d_in[1];
  const float* Wk = (const float*)d_in[2];
  const float* Wv = (const float*)d_in[3];
  const float* Wu = (const float*)d_in[4];
  const float* bu = (const float*)d_in[5];
  const float* W1 = (const float*)d_in[6];
  const float* b1 = (const float*)d_in[7];
  const float* W2 = (const float*)d_in[8];
  const float* b2 = (const float*)d_in[9];
  float* out = (float*)d_out;

  // workspace carve-up (~19 MB of fp32; L2-resident)
  float* w = (float*)d_ws;
  int*   rowmap  = (int*)w;       w += ROWS;
  float* hidden  = w;             w += (size_t)ROWS * 1024;
  float* params  = w;             w += (size_t)ROWS * 16;
  int*   indices = (int*)w;       w += (size_t)ROWS * VS;
  float* weights = w;             w += (size_t)ROWS * VS;
  float* qsel    = w;             w += (size_t)ROWS * 2048;
  float* qk      = w;             w += (size_t)ROWS * 2048;
  float* ctxb    = w;             w += (size_t)ROWS * 2048;
  float* ctxv    = w;             w += (size_t)ROWS * 2048;

  // jax.random.split(jax.random.key(42)): threefry((0,42), iota(4)) -> (2,2)
  unsigned kg0, kg1, kr0, kr1, o0a, o1a, o0b, o1b;
  threefry2x32(0u, 42u, 0u, 2u, &o0a, &o1a);
  threefry2x32(0u, 42u, 1u, 3u, &o0b, &o1b);
  kg0 = o0a; kg1 = o0b;   // keys[0]
  kr0 = o1a; kr1 = o1b;   // keys[1]

  init_rowmap<<<2, 256, 0, stream>>>(rowmap);
  fill_bias<<<(BB * TT * EMB) / 256, 256, 0, stream>>>(out, bu, BB * TT * EMB);
  mlp_hidden<<<ROWS, 256, 0, stream>>>(x, W1, b1, rowmap, hidden);
  mlp_params<<<ROWS, 256, 0, stream>>>(hidden, W2, b2, params);
  index_weight<<<ROWS, 64, 0, stream>>>(params, indices, weights, kg0, kg1, kr0, kr1);

  // qsel = x_sel @ Wq                  (512 x 2048, K=256) — gather-A
  wmma_gemm<false, true, false, false><<<512, 256, 0, stream>>>(
      x, EMB, 0, rowmap, Wq, 2048, 0, qsel, 2048, 0, nullptr, 1.0f,
      ROWS, 2048, EMB);
  // qk_h = qsel_h @ Wk_h^T / 16        (8 heads x 512 x 256, K=256) — BT
  wmma_gemm<true, false, false, false><<<dim3(64, 8), 256, 0, stream>>>(
      qsel, 2048, 256, rowmap, Wk, 2048, 256, qk, 2048, 256, nullptr,
      1.0f / 16.0f, ROWS, EMB, EMB);
  // dot / softmax / ctx gather
  attn_ctx<<<BB * TP * HEADS, 256, 0, stream>>>(x, qk, indices, weights, ctxb);
  // ctxv_h = ctx_h @ Wv_h              (8 heads x 512 x 256, K=256)
  wmma_gemm<false, false, false, false><<<dim3(64, 8), 256, 0, stream>>>(
      ctxb, EMB, (size_t)ROWS * EMB, rowmap, Wv, 2048, 256, ctxv, 2048, 256,
      nullptr, 1.0f, ROWS, EMB, EMB);
  // out_sel = ctxv @ Wu + bu           (512 x 256, K=2048) — scatter-C + bias
  wmma_gemm<false, false, true, true><<<64, 256, 0, stream>>>(
      ctxv, 2048, 0, rowmap, Wu, EMB, 0, out, EMB, 0, bu, 1.0f,
      ROWS, EMB, 2048);
}